// MaxAggregator_27376121545088
// MI455X (gfx1250) — compile-verified
//
#include <hip/hip_runtime.h>

// Segment-max over sorted segment ids (jax.ops.segment_max equivalent).
//   values:      [nRows, 64] f32  (streamed once -> non-temporal b128 loads)
//   segment_ids: [nRows] i32, sorted ascending (L2-resident: 20MB << 192MB)
//   out:         [numSegs, 64] f32
//
// One wave32 owns one segment: a *scalar* (SALU/SMEM) binary search gives the
// [s, e) row range; 16 lanes x float4 cover a 64-float row, the two wave
// halves cover rows r and r+1 (512 contiguous bytes per wave load). No
// atomics, no LDS staging (each byte used once). HBM roofline ~56us.

typedef __attribute__((ext_vector_type(4))) float v4f;

#define WAVE_SIZE        32
#define WAVES_PER_BLOCK  8
#define BLOCK_THREADS    (WAVE_SIZE * WAVES_PER_BLOCK)
#define FEAT             64
#define F4_PER_ROW       (FEAT / 4)   // 16 float4 groups per row

__device__ __forceinline__ v4f vmax4(v4f a, v4f b) {
    v4f r;
    r.x = fmaxf(a.x, b.x);
    r.y = fmaxf(a.y, b.y);
    r.z = fmaxf(a.z, b.z);
    r.w = fmaxf(a.w, b.w);
    return r;
}

// First index i in [0, n) with ids[i] >= key. All operands are wave-uniform
// (key comes from readfirstlane), so this lowers to SALU + s_load_b32.
__device__ __forceinline__ int lower_bound_u(const int* __restrict__ ids,
                                             int n, int key) {
    int lo = 0, hi = n;
    while (lo < hi) {
        int mid = (lo + hi) >> 1;
        if (ids[mid] < key) lo = mid + 1; else hi = mid;
    }
    return lo;
}

__global__ __launch_bounds__(BLOCK_THREADS)
void seg_max_f32_kernel(const float* __restrict__ values,
                        const int*   __restrict__ ids,
                        float*       __restrict__ out,
                        int nRows, int numSegs) {
    const int wave = threadIdx.x >> 5;
    const int lane = threadIdx.x & (WAVE_SIZE - 1);

    // seg is wave-uniform by construction; force it into an SGPR so the
    // binary search scalarizes (SALU + scalar loads, no per-lane VALU chain).
    const int seg = __builtin_amdgcn_readfirstlane(blockIdx.x * WAVES_PER_BLOCK + wave);
    if (seg >= numSegs) return;

    // Row range owned by this segment (exclusive ownership -> no atomics).
    const int s = lower_bound_u(ids, nRows, seg);
    const int e = lower_bound_u(ids, nRows, seg + 1);

    const int half = lane >> 4;   // 0: rows s+0,s+2,...  1: rows s+1,s+3,...
    const int f4   = lane & 15;   // which float4 of the 64-float row

    const float NEG_INF = -__builtin_huge_valf();
    v4f m = { NEG_INF, NEG_INF, NEG_INF, NEG_INF };

    int r = s + half;
    const v4f* __restrict__ p =
        (const v4f*)values + (size_t)r * F4_PER_ROW + f4;

    // Main loop: 8 rows per wave iteration, 4 NT b128 loads in flight per
    // lane at fixed byte offsets (0/512/1024/1536 -> IOFFSET immediates),
    // one pointer bump per iteration. Speculative global_prefetch_b8 at 8KB
    // distance (translation failures are silently dropped -> no guard).
    for (; r + 6 < e; r += 8) {
        __builtin_prefetch(p + 32 * F4_PER_ROW, 0, 0);
        v4f a0 = __builtin_nontemporal_load(p);
        v4f a1 = __builtin_nontemporal_load(p + 2 * F4_PER_ROW);
        v4f a2 = __builtin_nontemporal_load(p + 4 * F4_PER_ROW);
        v4f a3 = __builtin_nontemporal_load(p + 6 * F4_PER_ROW);
        m = vmax4(m, vmax4(vmax4(a0, a1), vmax4(a2, a3)));
        p += 8 * F4_PER_ROW;
    }
    // Tail rows (at most 3 stride-2 steps).
    for (; r < e; r += 2) {
        m = vmax4(m, __builtin_nontemporal_load(p));
        p += 2 * F4_PER_ROW;
    }

    // Combine the two 16-lane halves (wave32 shuffle; mask 16 swaps halves).
    m.x = fmaxf(m.x, __shfl_xor(m.x, 16, WAVE_SIZE));
    m.y = fmaxf(m.y, __shfl_xor(m.y, 16, WAVE_SIZE));
    m.z = fmaxf(m.z, __shfl_xor(m.z, 16, WAVE_SIZE));
    m.w = fmaxf(m.w, __shfl_xor(m.w, 16, WAVE_SIZE));

    // Lanes 0..15 hold the finished row -> one 128-bit NT store each.
    if (half == 0) {
        v4f* __restrict__ obase = (v4f*)out;
        __builtin_nontemporal_store(m, &obase[(size_t)seg * F4_PER_ROW + f4]);
    }
}

extern "C" void kernel_launch(void* const* d_in, const int* in_sizes, int n_in,
                              void* d_out, int out_size, void* d_ws, size_t ws_size,
                              hipStream_t stream) {
    const float* values = (const float*)d_in[0];
    const int*   ids    = (const int*)d_in[1];
    float*       out    = (float*)d_out;

    const int nRows   = in_sizes[1];         // segment_ids element count
    const int numSegs = out_size / FEAT;     // out is [numSegs, 64]

    const int blocks = (numSegs + WAVES_PER_BLOCK - 1) / WAVES_PER_BLOCK;
    seg_max_f32_kernel<<<blocks, BLOCK_THREADS, 0, stream>>>(
        values, ids, out, nRows, numSegs);
}